// LowGCN_88510686036815
// MI455X (gfx1250) — compile-verified
//
#include <hip/hip_runtime.h>
#include <hip/hip_bf16.h>

typedef __attribute__((ext_vector_type(2))) float v2f;
typedef __attribute__((ext_vector_type(8))) float v8f;

#define BN_EPS 1e-5f

// ---------------------------------------------------------------------------
// Degree / D^{-1/2}
// ---------------------------------------------------------------------------
__global__ void deg_kernel(const int* __restrict__ dst, float* __restrict__ deg, int E) {
    int i = blockIdx.x * blockDim.x + threadIdx.x;
    if (i < E) atomicAdd(&deg[dst[i]], 1.0f);
}

__global__ void dinv_kernel(float* __restrict__ deg, int N) {
    int i = blockIdx.x * blockDim.x + threadIdx.x;
    if (i < N) deg[i] = rsqrtf(fmaxf(deg[i], 1.0f));
}

// ---------------------------------------------------------------------------
// Edge scatter: agg[dst] += h[src] * dinv[src].  One wave32 per edge; each
// lane moves one float4 (32 lanes * 16B = full 512B row), then 4 f32 atomics.
// ---------------------------------------------------------------------------
__global__ __launch_bounds__(256) void scatter_edges(
    const float* __restrict__ H, const int* __restrict__ src,
    const int* __restrict__ dst, const float* __restrict__ dinv,
    float* __restrict__ agg, int E)
{
    int gw   = (int)((blockIdx.x * 256u + threadIdx.x) >> 5);
    int lane = threadIdx.x & 31;
    if (gw >= E) return;
    int s = src[gw];
    int d = dst[gw];
    float sc = dinv[s];
    float4 v = *(const float4*)(H + (size_t)s * 128 + lane * 4);
    float* o = agg + (size_t)d * 128 + lane * 4;
    atomicAdd(o + 0, v.x * sc);
    atomicAdd(o + 1, v.y * sc);
    atomicAdd(o + 2, v.z * sc);
    atomicAdd(o + 3, v.w * sc);
}

// h = h + agg * dinv[row]   (in place, float4 vectorized; 32 float4 per row)
__global__ void combine_kernel(float* __restrict__ h, const float* __restrict__ agg,
                               const float* __restrict__ dinv, int n4)
{
    int i = blockIdx.x * blockDim.x + threadIdx.x;
    if (i >= n4) return;
    int row = i >> 5;
    float s = dinv[row];
    float4 a = ((const float4*)agg)[i];
    float4 v = ((float4*)h)[i];
    v.x += a.x * s; v.y += a.y * s; v.z += a.z * s; v.w += a.w * s;
    ((float4*)h)[i] = v;
}

// ---------------------------------------------------------------------------
// WMMA fp32 GEMM:  C[M,NCOL] = A[M,K] @ W[K,NCOL] (+bias) (optional fused
// BN-affine+ReLU on A at tile-load).
//   * Block = 256 thr = 8 waves, owns a 32-row A panel in LDS (stride K+4 ->
//     conflict-free column reads, ds_load_2addr_b64 fragment fetch).
//   * Each wave owns a 16-column slab and stacks TWO 16x16 WMMA tiles in M,
//     so every B fragment (2 global loads) feeds two V_WMMA_F32_16X16X4_F32
//     with independent accumulator chains.
//   * No row guards: last block overlap-clamps r0 to M-32 and benignly
//     rewrites identical values (M is a multiple of 16, M >= 32).
// ---------------------------------------------------------------------------
template<int K, int NCOL, bool BIAS, bool BNRELU>
__global__ __launch_bounds__(256) void gemm_wmma_kernel(
    const float* __restrict__ A, const float* __restrict__ W,
    const float* __restrict__ bias,
    const float* __restrict__ bnscale, const float* __restrict__ bnshift,
    float* __restrict__ C, int M)
{
    constexpr int LDA = K + 4;               // 4-bank rotation per row
    __shared__ float tile[32 * LDA];

    int r0 = blockIdx.x * 32;
    if (r0 > M - 32) r0 = M - 32;            // overlap-clamp, no guards needed

    // cooperative, coalesced 32xK panel load (with fused BN+ReLU if requested)
    for (int idx = threadIdx.x; idx < 32 * K; idx += 256) {
        int row = idx / K;
        int col = idx - row * K;
        float v = A[(size_t)(r0 + row) * K + col];
        if (BNRELU) v = fmaxf(v * bnscale[col] + bnshift[col], 0.0f);
        tile[row * LDA + col] = v;
    }
    __syncthreads();

    const int lane = threadIdx.x & 31;
    const int wv   = threadIdx.x >> 5;
    const int m    = lane & 15;            // A row / B col within tile
    const int half = lane >> 4;            // 0: K{0,1}/M 0-7 ; 1: K{2,3}/M 8-15
    const int koff = half * 2;
    constexpr int NT = NCOL / 16;

    for (int ct = wv; ct < NT; ct += 8) {
        const int n = ct * 16 + m;
        v8f acc0 = {};
        v8f acc1 = {};
        const float* tA0 = tile + m * LDA + koff;
        const float* tA1 = tile + (m + 16) * LDA + koff;
        const float* Wp  = W + (size_t)koff * NCOL + n;
        #pragma unroll 4
        for (int k = 0; k < K; k += 4) {
            v2f a0, a1, b;
            a0.x = tA0[k];
            a0.y = tA0[k + 1];
            a1.x = tA1[k];
            a1.y = tA1[k + 1];
            b.x  = Wp[(size_t)k * NCOL];
            b.y  = Wp[(size_t)(k + 1) * NCOL];
            acc0 = __builtin_amdgcn_wmma_f32_16x16x4_f32(
                false, a0, false, b, (short)0, acc0, false, false);
            acc1 = __builtin_amdgcn_wmma_f32_16x16x4_f32(
                false, a1, false, b, (short)0, acc1, false, false);
        }
        float* Cp = C + (size_t)(r0 + half * 8) * NCOL + n;
        if constexpr (BIAS) {
            float bv = bias[n];
            #pragma unroll
            for (int i = 0; i < 8; ++i)
                Cp[(size_t)i * NCOL] = acc0[i] + bv;
            Cp += (size_t)16 * NCOL;
            #pragma unroll
            for (int i = 0; i < 8; ++i)
                Cp[(size_t)i * NCOL] = acc1[i] + bv;
        } else {
            #pragma unroll
            for (int i = 0; i < 8; ++i)
                Cp[(size_t)i * NCOL] = acc0[i];
            Cp += (size_t)16 * NCOL;
            #pragma unroll
            for (int i = 0; i < 8; ++i)
                Cp[(size_t)i * NCOL] = acc1[i];
        }
    }
}

// ---------------------------------------------------------------------------
// BatchNorm statistics over 50k rows x 256 cols
// ---------------------------------------------------------------------------
__global__ __launch_bounds__(256) void bn_reduce_kernel(
    const float* __restrict__ Z, float* __restrict__ sums,
    float* __restrict__ sumsq, int M)
{
    int col = threadIdx.x;     // 256 threads = 256 columns, coalesced rows
    float s = 0.0f, q = 0.0f;
    for (int r = blockIdx.x; r < M; r += gridDim.x) {
        float v = Z[(size_t)r * 256 + col];
        s += v; q += v * v;
    }
    atomicAdd(&sums[col], s);
    atomicAdd(&sumsq[col], q);
}

__global__ void bn_finalize_kernel(const float* __restrict__ sums,
                                   const float* __restrict__ sumsq,
                                   const float* __restrict__ gamma,
                                   const float* __restrict__ beta,
                                   float* __restrict__ scale,
                                   float* __restrict__ shift, float invM)
{
    int c = threadIdx.x;
    float mu  = sums[c] * invM;
    float var = fmaxf(sumsq[c] * invM - mu * mu, 0.0f);
    float s   = gamma[c] * rsqrtf(var + BN_EPS);
    scale[c]  = s;
    shift[c]  = beta[c] - mu * s;
}

// ---------------------------------------------------------------------------
static inline size_t al256(size_t x) { return (x + 255) & ~(size_t)255; }

extern "C" void kernel_launch(void* const* d_in, const int* in_sizes, int n_in,
                              void* d_out, int out_size, void* d_ws, size_t ws_size,
                              hipStream_t stream) {
    const float* feat  = (const float*)d_in[0];
    const int*   src   = (const int*)  d_in[1];
    const int*   dst   = (const int*)  d_in[2];
    const float* W0    = (const float*)d_in[3];
    const float* W1    = (const float*)d_in[4];
    const float* Wp1   = (const float*)d_in[5];
    const float* bp1   = (const float*)d_in[6];
    const float* gamma = (const float*)d_in[7];
    const float* beta  = (const float*)d_in[8];
    const float* Wp2   = (const float*)d_in[9];
    const float* bp2   = (const float*)d_in[10];
    float* out = (float*)d_out;

    const int N = in_sizes[0] / 512;   // 50000
    const int E = in_sizes[1];         // 800000

    // workspace layout
    char* w = (char*)d_ws;
    size_t o = 0;
    float* dinv  = (float*)(w + o); o += al256((size_t)N * 4);
    float* bufA  = (float*)(w + o); o += al256((size_t)N * 128 * 4);
    float* bufB  = (float*)(w + o); o += al256((size_t)N * 128 * 4);
    float* bufZ  = (float*)(w + o); o += al256((size_t)N * 256 * 4);
    float* sums  = (float*)(w + o); o += 256 * 4;
    float* sumsq = (float*)(w + o); o += 256 * 4;
    float* scale = (float*)(w + o); o += 256 * 4;
    float* shift = (float*)(w + o); o += 256 * 4;
    (void)ws_size; (void)n_in; (void)out_size;

    const int rowTiles = (N + 31) / 32;            // 1563 (overlap-clamped)
    const int n4       = N * 32;                   // float4 elems of [N,128]

    // degrees -> D^{-1/2}
    hipMemsetAsync(dinv, 0, (size_t)N * 4, stream);
    deg_kernel<<<(E + 255) / 256, 256, 0, stream>>>(dst, dinv, E);
    dinv_kernel<<<(N + 255) / 256, 256, 0, stream>>>(dinv, N);

    // h0 = feat @ W0 ; conv0
    gemm_wmma_kernel<512, 128, false, false><<<rowTiles, 256, 0, stream>>>(
        feat, W0, nullptr, nullptr, nullptr, bufA, N);
    hipMemsetAsync(bufB, 0, (size_t)N * 128 * 4, stream);
    scatter_edges<<<(E + 7) / 8, 256, 0, stream>>>(bufA, src, dst, dinv, bufB, E);
    combine_kernel<<<(n4 + 255) / 256, 256, 0, stream>>>(bufA, bufB, dinv, n4);

    // h1 = h0 @ W1 ; conv1
    gemm_wmma_kernel<128, 128, false, false><<<rowTiles, 256, 0, stream>>>(
        bufA, W1, nullptr, nullptr, nullptr, bufB, N);
    hipMemsetAsync(bufA, 0, (size_t)N * 128 * 4, stream);
    scatter_edges<<<(E + 7) / 8, 256, 0, stream>>>(bufB, src, dst, dinv, bufA, E);
    combine_kernel<<<(n4 + 255) / 256, 256, 0, stream>>>(bufB, bufA, dinv, n4);

    // z = h1 @ Wp1 + bp1
    gemm_wmma_kernel<128, 256, true, false><<<rowTiles, 256, 0, stream>>>(
        bufB, Wp1, bp1, nullptr, nullptr, bufZ, N);

    // BatchNorm training-mode stats (biased var)
    hipMemsetAsync(sums, 0, 2 * 256 * 4, stream);
    bn_reduce_kernel<<<512, 256, 0, stream>>>(bufZ, sums, sumsq, N);
    bn_finalize_kernel<<<1, 256, 0, stream>>>(sums, sumsq, gamma, beta,
                                              scale, shift, 1.0f / (float)N);

    // out = relu(BN(z)) @ Wp2 + bp2   (BN+ReLU fused into A-tile load)
    gemm_wmma_kernel<256, 128, true, true><<<rowTiles, 256, 0, stream>>>(
        bufZ, Wp2, bp2, scale, shift, out, N);
}